// CausalSelfAttention_48911087567107
// MI455X (gfx1250) — compile-verified
//
#include <hip/hip_runtime.h>
#include <math.h>

// ---------------------------------------------------------------------------
// Problem constants (match reference)
// ---------------------------------------------------------------------------
#define BB 2
#define TT 512
#define CC 768
#define HH 12
#define HS 64
#define C3 2304      // 3*C
#define MTOP 8
#define NCOMB 36     // C(8,1) + C(8,2)

typedef __attribute__((ext_vector_type(2))) float v2f;
typedef __attribute__((ext_vector_type(8))) float v8f;

static __device__ __forceinline__ v8f wmma_f32(v2f a, v2f b, v8f c) {
    return __builtin_amdgcn_wmma_f32_16x16x4_f32(
        /*neg_a=*/false, a, /*neg_b=*/false, b,
        /*c_mod=*/(short)0, c, /*reuse_a=*/false, /*reuse_b=*/false);
}

// ---------------------------------------------------------------------------
// FP32 WMMA GEMM:  out[M,N] = A[M,K] @ W[N,K]^T + bias[N]
// One wave32 per 16x64 output tile: A-frag loaded once per K-step and reused
// across 4 B-frags / 4 accumulators. Software-pipelined: next K-step's five
// fragments are in flight while the current step's 4 WMMAs execute, so the
// wave overlaps VMEM latency with the matrix pipe instead of draining
// loadcnt to 0 before each WMMA. Requires K % 4 == 0, K >= 8.
// ---------------------------------------------------------------------------
__global__ __launch_bounds__(32)
void gemm_nt_bias_wmma(const float* __restrict__ A,
                       const float* __restrict__ W,     // [N,K] row-major (used transposed)
                       const float* __restrict__ bias,  // [N]
                       float* __restrict__ out,
                       int Mtot, int N, int K) {
    const int tm   = blockIdx.x * 16;
    const int tn   = blockIdx.y * 64;
    const int lane = threadIdx.x;
    const int half = lane >> 4;       // 0: lanes 0-15, 1: lanes 16-31
    const int l16  = lane & 15;

    v8f acc0 = {}, acc1 = {}, acc2 = {}, acc3 = {};

    const float* arow = A + (size_t)(tm + l16) * K;   // A-frag: m = l16
    const float* b0   = W + (size_t)(tn + l16) * K;   // B-frags: n = l16 within sub-tile
    const float* b1   = b0 + (size_t)16 * K;
    const float* b2   = b0 + (size_t)32 * K;
    const float* b3   = b0 + (size_t)48 * K;

    // ---- prologue: fragments for k0 = 0 -----------------------------------
    int ka = half * 2;                                // lanes 16-31 hold K+2,K+3
    v2f a  = { arow[ka], arow[ka + 1] };
    v2f w0 = { b0[ka],   b0[ka + 1]   };
    v2f w1 = { b1[ka],   b1[ka + 1]   };
    v2f w2 = { b2[ka],   b2[ka + 1]   };
    v2f w3 = { b3[ka],   b3[ka + 1]   };

    for (int k0 = 0; k0 < K - 4; k0 += 4) {
        // one prefetch per 128B line per stream, ~3 lines ahead (uniform branch)
        if ((k0 & 28) == 0) {
            __builtin_prefetch(arow + k0 + 96, 0, 3);
            __builtin_prefetch(b0   + k0 + 96, 0, 3);
            __builtin_prefetch(b1   + k0 + 96, 0, 3);
            __builtin_prefetch(b2   + k0 + 96, 0, 3);
            __builtin_prefetch(b3   + k0 + 96, 0, 3);
        }
        // issue next step's loads before this step's WMMAs
        const int kn = k0 + 4 + half * 2;
        v2f na  = { arow[kn], arow[kn + 1] };
        v2f nw0 = { b0[kn],   b0[kn + 1]   };
        v2f nw1 = { b1[kn],   b1[kn + 1]   };
        v2f nw2 = { b2[kn],   b2[kn + 1]   };
        v2f nw3 = { b3[kn],   b3[kn + 1]   };

        acc0 = wmma_f32(a, w0, acc0);
        acc1 = wmma_f32(a, w1, acc1);
        acc2 = wmma_f32(a, w2, acc2);
        acc3 = wmma_f32(a, w3, acc3);

        a = na; w0 = nw0; w1 = nw1; w2 = nw2; w3 = nw3;
    }
    // ---- epilogue: last K-step --------------------------------------------
    acc0 = wmma_f32(a, w0, acc0);
    acc1 = wmma_f32(a, w1, acc1);
    acc2 = wmma_f32(a, w2, acc2);
    acc3 = wmma_f32(a, w3, acc3);

    const float bv0 = bias[tn      + l16];
    const float bv1 = bias[tn + 16 + l16];
    const float bv2 = bias[tn + 32 + l16];
    const float bv3 = bias[tn + 48 + l16];
    #pragma unroll
    for (int r = 0; r < 8; ++r) {
        const int m = r + 8 * half;                   // C/D layout: VGPR r -> M=r (+8 hi half)
        float* orow = out + (size_t)(tm + m) * N + tn + l16;
        orow[0]  = acc0[r] + bv0;
        orow[16] = acc1[r] + bv1;
        orow[32] = acc2[r] + bv2;
        orow[48] = acc3[r] + bv3;
    }
}

// ---------------------------------------------------------------------------
// Per-(b,h,t) DPP token kernel — one wave32 per token.
// qkv layout: [B,T,3C] with q at col 0, k at col 768, v at col 1536, head h
// occupying cols h*64..h*64+63 inside each third.
// ---------------------------------------------------------------------------
__global__ __launch_bounds__(32)
void dpp_token_kernel(const float* __restrict__ qkv, float* __restrict__ y) {
    const int t    = blockIdx.x;
    const int h    = blockIdx.y;
    const int b    = blockIdx.z;
    const int lane = threadIdx.x;

    __shared__ float qls[HS];
    __shared__ float sim[TT];
    __shared__ float Kc[9][HS];
    __shared__ float Vc[9][HS];
    __shared__ float gram[9][9];
    __shared__ float qd[9];
    __shared__ int   topidx[MTOP];
    __shared__ float scores_s[NCOMB];
    __shared__ float wv_s[NCOMB][3];
    __shared__ int   wr_s[NCOMB][3];
    __shared__ float coef[9];

    const size_t rowbase = (size_t)(b * TT + t) * C3;
    const float* qrow = qkv + rowbase + h * HS;

    // ---- Phase A: load q_i, compute causal similarities -------------------
    qls[lane]      = qrow[lane];
    qls[lane + 32] = qrow[lane + 32];
    __syncthreads();

    for (int tt = lane; tt <= t; tt += 32) {
        const float* kr = qkv + (size_t)(b * TT + tt) * C3 + CC + h * HS;
        float s = 0.f;
        #pragma unroll 8
        for (int d = 0; d < HS; ++d) s += kr[d] * qls[d];
        sim[tt] = s;
    }
    __syncthreads();

    // ---- Phase B: serial top-8 via wave argmax ----------------------------
    unsigned selmask = 0u;   // bit j -> candidate t = lane + 32*j already taken
    for (int m = 0; m < MTOP; ++m) {
        float bestv = -INFINITY;
        int   besti = 0x7fffffff;
        int   j = 0;
        for (int tt = lane; tt <= t; tt += 32, ++j) {
            if (!((selmask >> j) & 1u)) {
                float v = sim[tt];
                if (v > bestv || (v == bestv && tt < besti)) { bestv = v; besti = tt; }
            }
        }
        #pragma unroll
        for (int off = 16; off > 0; off >>= 1) {
            float ov = __shfl_xor(bestv, off, 32);
            int   oi = __shfl_xor(besti, off, 32);
            if (ov > bestv || (ov == bestv && oi < besti)) { bestv = ov; besti = oi; }
        }
        if (lane == 0) topidx[m] = (besti == 0x7fffffff) ? 0 : besti;
        if (besti != 0x7fffffff && (besti & 31) == lane) selmask |= 1u << (besti >> 5);
    }
    __syncthreads();

    // ---- Phase C: stage candidate K/V rows (row 0 = token i) --------------
    for (int r = 0; r < 9; ++r) {
        const int idx = (r == 0) ? t : topidx[r - 1];
        const float* kr = qkv + (size_t)(b * TT + idx) * C3 + CC + h * HS;
        const float* vr = kr + CC;   // v is 768 past k
        Kc[r][lane]      = kr[lane];
        Kc[r][lane + 32] = kr[lane + 32];
        Vc[r][lane]      = vr[lane];
        Vc[r][lane + 32] = vr[lane + 32];
    }
    __syncthreads();

    // ---- qdot[r] = <K_r, q_i> / sqrt(HS) ----------------------------------
    for (int r = 0; r < 9; ++r) {
        float p = Kc[r][lane] * qls[lane] + Kc[r][lane + 32] * qls[lane + 32];
        #pragma unroll
        for (int off = 16; off > 0; off >>= 1) p += __shfl_xor(p, off, 32);
        if (lane == 0) qd[r] = p * 0.125f;   // 1/sqrt(64)
    }
    // ---- 9x9 Gram matrix: 45 unique pairs, lane-parallel ------------------
    for (int base = 0; base < 45; base += 32) {
        const int p = base + lane;
        if (p < 45) {
            int r1 = 0, rem = p;
            while (rem >= 9 - r1) { rem -= 9 - r1; ++r1; }
            const int r2 = r1 + rem;
            float s = 0.f;
            #pragma unroll 8
            for (int d = 0; d < HS; ++d) s += Kc[r1][d] * Kc[r2][d];
            gram[r1][r2] = s;
            gram[r2][r1] = s;
        }
    }
    __syncthreads();

    // ---- Phase D: 36 combos, lane-parallel --------------------------------
    const int ncand = (t + 1 < MTOP) ? (t + 1) : MTOP;
    for (int c = lane; c < NCOMB; c += 32) {
        int s, sl0, sl1 = -1;
        if (c < 8) { s = 1; sl0 = c; }
        else {
            int c2 = c - 8, p = 0, rem = c2;
            while (rem >= 7 - p) { rem -= 7 - p; ++p; }
            s = 2; sl0 = p; sl1 = p + 1 + rem;
        }
        bool valid = (sl0 < ncand) && (topidx[sl0] != t);
        if (s == 2) valid = valid && (sl1 < ncand) && (topidx[sl1] != t);

        const int ra = sl0 + 1;
        const int rb = (s == 2) ? sl1 + 1 : 0;

        float det;
        if (s == 1) {
            det = gram[0][0] * gram[ra][ra] - gram[0][ra] * gram[0][ra];
        } else {
            const float g00 = gram[0][0],  g01 = gram[0][ra], g02 = gram[0][rb];
            const float g11 = gram[ra][ra], g12 = gram[ra][rb], g22 = gram[rb][rb];
            det = g00 * (g11 * g22 - g12 * g12)
                - g01 * (g01 * g22 - g12 * g02)
                + g02 * (g01 * g12 - g11 * g02);
        }
        const float ld = logf((valid ? det : 1.0f) + 1e-6f);
        scores_s[c] = valid ? (ld / (float)(s + 1)) : -1e30f;

        // subset softmax weights over rows {0, ra, (rb)}
        const float d0 = qd[0], da = qd[ra];
        const float db = (s == 2) ? qd[rb] : -INFINITY;
        float mx = fmaxf(d0, da);
        if (s == 2) mx = fmaxf(mx, db);
        const float e0 = expf(d0 - mx);
        const float ea = expf(da - mx);
        const float eb = (s == 2) ? expf(db - mx) : 0.f;
        const float inv = 1.f / (e0 + ea + eb);
        wr_s[c][0] = 0;  wv_s[c][0] = e0 * inv;
        wr_s[c][1] = ra; wv_s[c][1] = ea * inv;
        wr_s[c][2] = rb; wv_s[c][2] = (s == 2) ? eb * inv : 0.f;
    }
    if (lane < 9) coef[lane] = 0.f;
    __syncthreads();

    // ---- Phase E: softmax over combo scores; accumulate row coefficients --
    float lmax = -INFINITY;
    for (int c = lane; c < NCOMB; c += 32) lmax = fmaxf(lmax, scores_s[c]);
    #pragma unroll
    for (int off = 16; off > 0; off >>= 1) lmax = fmaxf(lmax, __shfl_xor(lmax, off, 32));
    float lsum = 0.f;
    for (int c = lane; c < NCOMB; c += 32) lsum += expf(scores_s[c] - lmax);
    #pragma unroll
    for (int off = 16; off > 0; off >>= 1) lsum += __shfl_xor(lsum, off, 32);
    const bool has_valid = lmax > -5e29f;

    for (int c = lane; c < NCOMB; c += 32) {
        const float pc = expf(scores_s[c] - lmax) / lsum;
        #pragma unroll
        for (int j = 0; j < 3; ++j)
            atomicAdd(&coef[wr_s[c][j]], pc * wv_s[c][j]);
    }
    __syncthreads();

    // ---- Phase F: y = sum_r coef[r] * V_r  (fallback v[i]) ----------------
    float* yrow = y + (size_t)(b * TT + t) * CC + h * HS;
    for (int d = lane; d < HS; d += 32) {
        float acc = 0.f;
        #pragma unroll
        for (int r = 0; r < 9; ++r) acc += coef[r] * Vc[r][d];
        yrow[d] = has_valid ? acc : Vc[0][d];
    }
}

// ---------------------------------------------------------------------------
// Launch
// ---------------------------------------------------------------------------
extern "C" void kernel_launch(void* const* d_in, const int* in_sizes, int n_in,
                              void* d_out, int out_size, void* d_ws, size_t ws_size,
                              hipStream_t stream) {
    const float* x      = (const float*)d_in[0];   // [B,T,C]
    const float* W_attn = (const float*)d_in[1];   // [3C,C]
    const float* b_attn = (const float*)d_in[2];   // [3C]
    const float* W_proj = (const float*)d_in[3];   // [C,C]
    const float* b_proj = (const float*)d_in[4];   // [C]
    float* out = (float*)d_out;                    // [B,T,C]

    const int Mtot = BB * TT;                      // 1024
    float* qkv = (float*)d_ws;                                     // 1024*2304 f32
    float* ybf = (float*)((char*)d_ws + (size_t)Mtot * C3 * 4);    // 1024*768  f32

    // qkv = x @ W_attn^T + b_attn       (N tiles of 64)
    dim3 gq(Mtot / 16, C3 / 64);
    gemm_nt_bias_wmma<<<gq, 32, 0, stream>>>(x, W_attn, b_attn, qkv, Mtot, C3, CC);

    // per-token DPP attention
    dim3 gt(TT, HH, BB);
    dpp_token_kernel<<<gt, 32, 0, stream>>>(qkv, ybf);

    // out = y @ W_proj^T + b_proj
    dim3 gp(Mtot / 16, CC / 64);
    gemm_nt_bias_wmma<<<gp, 32, 0, stream>>>(ybf, W_proj, b_proj, out, Mtot, CC, CC);
}